// CausalSelfAttention_15547781611845
// MI455X (gfx1250) — compile-verified
//
#include <hip/hip_runtime.h>

// ---------------------------------------------------------------------------
// Causal self-attention, fp32, gfx1250 (MI455X).
//  Stage 1: qkv = x @ w_qkv + b_qkv   (TDM-staged block GEMM, WMMA f32)
//  Stage 2: flash-style attention, 4 waves/block sharing TDM-staged K/V tiles
//  Stage 3: out = y @ w_proj + b_proj (same GEMM)
// ---------------------------------------------------------------------------

typedef __attribute__((ext_vector_type(2))) float v2f;
typedef __attribute__((ext_vector_type(8))) float v8f;
typedef __attribute__((ext_vector_type(4))) unsigned int u32x4;
typedef __attribute__((ext_vector_type(8))) int i32x8;
typedef __attribute__((ext_vector_type(4))) int i32x4;

#define B_DIM   2
#define T_DIM   2048
#define C_DIM   1024
#define H_DIM   64
#define N_HEADS 16
#define QKV_N   3072

__device__ __forceinline__ v8f wmma4(v2f a, v2f b, v8f c) {
  return __builtin_amdgcn_wmma_f32_16x16x4_f32(
      false, a, false, b, (short)0, c, false, false);
}

// ---------------------------------------------------------------------------
// TDM: DMA one 2D fp32 tile (tile_w x tile_h, row stride stride_elems) from
// global memory into LDS, inserting (pad_amount_code+1) DWORDs of padding
// after every 2^(pad_interval_code+1) DWORDs (sized = one tile row).
// D# layout per CDNA5 ISA 8.3/8.4.  Tracked on TENSORcnt.
// 6-arg builtin on this toolchain:
//   (u32x4 g0, i32x8 g1, i32x4 g2, i32x4 g3, i32x8 extra, i32 cpol)
// ---------------------------------------------------------------------------
__device__ __forceinline__ void tdm_load_tile_f32(
    unsigned lds_byte_off, const float* g, unsigned tile_w, unsigned tile_h,
    unsigned stride_elems, unsigned pad_interval_code, unsigned pad_amount_code) {
  unsigned long long ga = (unsigned long long)(uintptr_t)g;
  const unsigned td0 = stride_elems;   // row length -> never OOB inside a row
  const unsigned td1 = tile_h;

  u32x4 g0;
  g0[0] = 1u;                                   // count=1 (valid), no gather
  g0[1] = lds_byte_off;                         // lds_addr
  g0[2] = (unsigned)ga;                         // global_addr[31:0]
  g0[3] = ((unsigned)(ga >> 32) & 0x01ffffffu)  // global_addr[56:32]
          | (2u << 30);                         // type = 2 (image)

  i32x8 g1;
  g1[0] = (int)((2u << 16)                      // data_size = 4 bytes
                | (1u << 20)                    // pad_enable
                | (pad_interval_code << 22)
                | (pad_amount_code << 25));
  g1[1] = (int)((td0 & 0xffffu) << 16);                     // dim0[15:0]
  g1[2] = (int)((td0 >> 16) | ((td1 & 0xffffu) << 16));     // dim0 hi | dim1 lo
  g1[3] = (int)((td1 >> 16) | (tile_w << 16));              // dim1 hi | tile_dim0
  g1[4] = (int)(tile_h & 0xffffu);                          // tile_dim1, tile_dim2=0
  g1[5] = (int)stride_elems;                                // dim0_stride[31:0]
  g1[6] = 0;
  g1[7] = 0;

  i32x4 gz = {0, 0, 0, 0};                      // 2D: groups 2/3 unused
  i32x8 z8 = {0, 0, 0, 0, 0, 0, 0, 0};
  __builtin_amdgcn_tensor_load_to_lds(g0, g1, gz, gz, z8, 0);
}

// ---------------------------------------------------------------------------
// Block GEMM: C[M,N] = A[M,K] @ W[K,N] + bias[N]
// 256 threads = 8 waves; block tile 128x128; wave tile 64x32 (4x2 WMMA tiles).
// K staged 16 at a time through double-buffered LDS, filled by the TDM.
// ---------------------------------------------------------------------------
#define BM 128
#define BN 128
#define BK 16
#define LDA_STRIDE 18    // 16 + 2 pad (pad_interval code 3 = 16 DW)
#define LDB_STRIDE 130   // 128 + 2 pad (pad_interval code 6 = 128 DW)

__global__ __launch_bounds__(256)
void gemm_bias_tdm(const float* __restrict__ A, const float* __restrict__ W,
                   const float* __restrict__ bias, float* __restrict__ Cout,
                   int M, int N, int K) {
  __shared__ float sA[2][BM * LDA_STRIDE];
  __shared__ float sB[2][BK * LDB_STRIDE];

  const int tid  = threadIdx.x;
  const int wid  = tid >> 5;
  const int lane = tid & 31;
  const int half = lane >> 4;
  const int lm   = lane & 15;
  const int wr   = wid >> 2;           // 0..1 : 64-row slice
  const int wc   = wid & 3;            // 0..3 : 32-col slice
  const int m0   = blockIdx.y * BM;
  const int n0   = blockIdx.x * BN;
  const int kSteps = K >> 4;

  auto issue = [&](int s) {
    const int buf = s & 1;
    tdm_load_tile_f32((unsigned)(uintptr_t)&sA[buf][0],
                      A + (size_t)m0 * K + s * BK, BK, BM, K, 3, 1);
    tdm_load_tile_f32((unsigned)(uintptr_t)&sB[buf][0],
                      W + (size_t)(s * BK) * N + n0, BN, BK, N, 6, 1);
  };

  if (wid == 0) {                       // wave 0 drives the TDM pipeline
    issue(0);
    if (kSteps > 1) issue(1);
  }

  v8f zero = {};
  v8f acc[4][2];
#pragma unroll
  for (int i = 0; i < 4; ++i)
#pragma unroll
    for (int j = 0; j < 2; ++j) acc[i][j] = zero;

  for (int s = 0; s < kSteps; ++s) {
    const int buf = s & 1;
    if (wid == 0) {
      if (s + 1 < kSteps) __builtin_amdgcn_s_wait_tensorcnt(2);
      else                __builtin_amdgcn_s_wait_tensorcnt(0);
    }
    __syncthreads();

    const float* __restrict__ a_ = &sA[buf][0];
    const float* __restrict__ b_ = &sB[buf][0];
#pragma unroll
    for (int kb = 0; kb < 4; ++kb) {
      v2f af[4];
#pragma unroll
      for (int i = 0; i < 4; ++i) {
        const int row = 64 * wr + 16 * i + lm;
        af[i] = *(const v2f*)(a_ + row * LDA_STRIDE + 4 * kb + 2 * half);
      }
      v2f bf[2];
#pragma unroll
      for (int j = 0; j < 2; ++j) {
        const int col = 32 * wc + 16 * j + lm;
        const int kr  = 4 * kb + 2 * half;
        bf[j].x = b_[kr * LDB_STRIDE + col];
        bf[j].y = b_[(kr + 1) * LDB_STRIDE + col];
      }
#pragma unroll
      for (int i = 0; i < 4; ++i)
#pragma unroll
        for (int j = 0; j < 2; ++j)
          acc[i][j] = wmma4(af[i], bf[j], acc[i][j]);
    }
    __syncthreads();
    if (wid == 0 && s + 2 < kSteps) issue(s + 2);
  }

#pragma unroll
  for (int j = 0; j < 2; ++j) {
    const int col = n0 + 32 * wc + 16 * j + lm;
    const float bn = bias[col];
#pragma unroll
    for (int i = 0; i < 4; ++i) {
#pragma unroll
      for (int r = 0; r < 8; ++r) {
        const int row = m0 + 64 * wr + 16 * i + r + 8 * half;
        Cout[(size_t)row * N + col] = acc[i][j][r] + bn;
      }
    }
  }
}

// ---------------------------------------------------------------------------
// Flash-style causal attention.
// 128 threads = 4 waves; wave w owns q-tile (blockIdx.x*4 + w) of 16 rows.
// K/V 16x64 tiles are TDM-staged into double-buffered LDS (stride 66 via DMA
// padding: 64-DW rows -> pad_interval code 5, +2 DW) and shared by all waves.
// All waves iterate kt = 0..qtmax so barriers stay uniform; compute is gated
// by the wave-uniform causal bound kt <= qt.  Per-wave softmax goes through a
// private LDS slice; same-wave LDS ops are in-order (no block barrier needed).
// ---------------------------------------------------------------------------
#define KV_STRIDE 66     // 64 + 2 pad

__global__ __launch_bounds__(128)
void attn_kernel(const float* __restrict__ qkv, float* __restrict__ Y) {
  const int tid  = threadIdx.x;
  const int wid  = tid >> 5;           // 0..3
  const int lane = tid & 31;
  const int half = lane >> 4;
  const int lm   = lane & 15;
  const int qt   = blockIdx.x * 4 + wid;
  const int q0   = qt * 16;
  const int h    = blockIdx.y;
  const int b    = blockIdx.z;
  const int nkt  = blockIdx.x * 4 + 4;   // kv tiles needed by the last wave

  __shared__ float sK[2][16 * KV_STRIDE];
  __shared__ float sV[2][16 * KV_STRIDE];
  __shared__ float sS[4][16 * 17];       // per-wave score/prob tile
  __shared__ float sAux[4][16];          // per-wave alpha / row-sum broadcast

  const float* __restrict__ qbase =
      qkv + (size_t)b * T_DIM * QKV_N + h * H_DIM;
  const float* __restrict__ kbase = qbase + C_DIM;
  const float* __restrict__ vbase = qbase + 2 * C_DIM;

  auto issue = [&](int kt) {
    const int buf = kt & 1;
    const float* ksrc = kbase + (size_t)(kt * 16) * QKV_N;
    const float* vsrc = vbase + (size_t)(kt * 16) * QKV_N;
    tdm_load_tile_f32((unsigned)(uintptr_t)&sK[buf][0], ksrc, 64, 16, QKV_N, 5, 1);
    tdm_load_tile_f32((unsigned)(uintptr_t)&sV[buf][0], vsrc, 64, 16, QKV_N, 5, 1);
  };

  if (wid == 0) {
    issue(0);
    if (nkt > 1) issue(1);
  }

  // Q tile (16x64) as 16 A-fragments, resident in VGPRs.
  v2f qa[16];
  {
    const float* __restrict__ qrow =
        qbase + (size_t)(q0 + lm) * QKV_N + 2 * half;
#pragma unroll
    for (int kb = 0; kb < 16; ++kb)
      qa[kb] = *(const v2f*)(qrow + 4 * kb);
  }

  v8f o[4] = {{}, {}, {}, {}};
  float mrow = -3.0e38f;
  float lrow = 0.0f;
  const float scale = 0.125f;       // 1/sqrt(64)
  float* __restrict__ myS = &sS[wid][0];
  float* __restrict__ myAux = &sAux[wid][0];

  for (int kt = 0; kt < nkt; ++kt) {
    const int buf = kt & 1;
    if (wid == 0) {
      if (kt + 1 < nkt) __builtin_amdgcn_s_wait_tensorcnt(2);
      else              __builtin_amdgcn_s_wait_tensorcnt(0);
    }
    __syncthreads();

    if (kt <= qt) {                      // wave-uniform causal gate
      const float* __restrict__ k_ = &sK[buf][0];
      const float* __restrict__ v_ = &sV[buf][0];
      const int kk0 = kt * 16;

      // ---- S = Q @ K^T --------------------------------------------------
      v8f s = {};
#pragma unroll
      for (int kb = 0; kb < 16; ++kb) {
        v2f bf = *(const v2f*)(k_ + lm * KV_STRIDE + 4 * kb + 2 * half);
        s = wmma4(qa[kb], bf, s);
      }

      // ---- scale + causal mask, spill to this wave's LDS slice ----------
      const bool diag = (kt == qt);
#pragma unroll
      for (int r = 0; r < 8; ++r) {
        const int qrow = q0 + r + 8 * half;
        const int kcol = kk0 + lm;
        float v = s[r] * scale;
        if (diag && kcol > qrow) v = -3.0e38f;
        myS[(r + 8 * half) * 17 + lm] = v;
      }

      // ---- online softmax: lanes 0..15 own one row each -----------------
      if (lane < 16) {
        float mx = myS[lane * 17];
#pragma unroll
        for (int j = 1; j < 16; ++j) mx = fmaxf(mx, myS[lane * 17 + j]);
        const float mnew  = fmaxf(mrow, mx);
        const float alpha = __expf(mrow - mnew);
        float sum = 0.0f;
#pragma unroll
        for (int j = 0; j < 16; ++j) {
          const float p = __expf(myS[lane * 17 + j] - mnew);
          myS[lane * 17 + j] = p;
          sum += p;
        }
        mrow = mnew;
        lrow = alpha * lrow + sum;
        myAux[lane] = alpha;
      }

      // ---- rescale O ----------------------------------------------------
      float al[8];
#pragma unroll
      for (int r = 0; r < 8; ++r) al[r] = myAux[r + 8 * half];
#pragma unroll
      for (int t = 0; t < 4; ++t)
#pragma unroll
        for (int r = 0; r < 8; ++r) o[t][r] *= al[r];

      // ---- P fragments --------------------------------------------------
      v2f pa[4];
#pragma unroll
      for (int kb = 0; kb < 4; ++kb) {
        pa[kb].x = myS[lm * 17 + 4 * kb + 2 * half];
        pa[kb].y = myS[lm * 17 + 4 * kb + 2 * half + 1];
      }

      // ---- O += P @ V ---------------------------------------------------
#pragma unroll
      for (int t = 0; t < 4; ++t) {
#pragma unroll
        for (int kb = 0; kb < 4; ++kb) {
          const int kr = 4 * kb + 2 * half;
          v2f vb;
          vb.x = v_[kr * KV_STRIDE + 16 * t + lm];
          vb.y = v_[(kr + 1) * KV_STRIDE + 16 * t + lm];
          o[t] = wmma4(pa[kb], vb, o[t]);
        }
      }
    }

    __syncthreads();
    if (wid == 0 && kt + 2 < nkt) issue(kt + 2);
  }

  // ---- finalize: divide by row sums, write Y ----------------------------
  if (lane < 16) myAux[lane] = lrow;
  float inv[8];
#pragma unroll
  for (int r = 0; r < 8; ++r) inv[r] = 1.0f / myAux[r + 8 * half];

  float* __restrict__ ybase = Y + (size_t)b * T_DIM * C_DIM + h * H_DIM;
#pragma unroll
  for (int t = 0; t < 4; ++t) {
#pragma unroll
    for (int r = 0; r < 8; ++r) {
      const int row = q0 + r + 8 * half;
      ybase[(size_t)row * C_DIM + t * 16 + lm] = o[t][r] * inv[r];
    }
  }
}

// ---------------------------------------------------------------------------
extern "C" void kernel_launch(void* const* d_in, const int* in_sizes, int n_in,
                              void* d_out, int out_size, void* d_ws,
                              size_t ws_size, hipStream_t stream) {
  (void)in_sizes; (void)n_in; (void)out_size; (void)ws_size;
  const float* x      = (const float*)d_in[0];
  const float* w_qkv  = (const float*)d_in[1];
  const float* b_qkv  = (const float*)d_in[2];
  const float* w_proj = (const float*)d_in[3];
  const float* b_proj = (const float*)d_in[4];
  float* out = (float*)d_out;

  float* qkv  = (float*)d_ws;                           // 50.3 MB
  float* yatt = qkv + (size_t)B_DIM * T_DIM * QKV_N;    // 16.8 MB

  const int M = B_DIM * T_DIM;  // 4096

  // Stage 1: QKV projection (M=4096, N=3072, K=1024)
  gemm_bias_tdm<<<dim3(QKV_N / BN, M / BM), dim3(256), 0, stream>>>(
      x, w_qkv, b_qkv, qkv, M, QKV_N, C_DIM);

  // Stage 2: causal attention (4 q-tiles per block share K/V staging)
  attn_kernel<<<dim3(T_DIM / 64, N_HEADS, B_DIM), dim3(128), 0, stream>>>(qkv, yatt);

  // Stage 3: output projection (M=4096, N=1024, K=1024)
  gemm_bias_tdm<<<dim3(C_DIM / BN, M / BM), dim3(256), 0, stream>>>(
      yatt, w_proj, b_proj, out, M, C_DIM, C_DIM);
}